// MultiLoss_80401787781717
// MI455X (gfx1250) — compile-verified
//
#include <hip/hip_runtime.h>
#include <math.h>

#define HIGH_THR_F 0.9f
#define NEGPOS 3
#define MAXG 128

typedef __attribute__((ext_vector_type(2))) float v2f;
typedef __attribute__((ext_vector_type(8))) float v8f;

// Wave-wide f32 sum using V_WMMA_F32_16X16X4_F32.
// A[m][k] layout (32-bit A 16x4): lanes 0-15 hold (M=lane, K=0/1) in vgpr0/1,
// lanes 16-31 hold (M=lane-16, K=2/3). We put the value in the first A vgpr and
// 0 in the second; B = all-ones (layout-invariant), so D[m][n] = v(m) + v(m+16).
// Lane-local sum of the 8 D vgprs gives the half-row-sum; xor-16 completes it.
__device__ __forceinline__ float wave_sum_wmma(float v) {
    v2f a; a[0] = v;    a[1] = 0.0f;
    v2f b; b[0] = 1.0f; b[1] = 1.0f;
    v8f c = {};
    v8f d = __builtin_amdgcn_wmma_f32_16x16x4_f32(false, a, false, b,
                                                  (short)0, c, false, false);
    float s = d[0] + d[1] + d[2] + d[3] + d[4] + d[5] + d[6] + d[7];
    s += __shfl_xor(s, 16, 32);
    return s;  // full 32-lane sum, broadcast to all lanes
}

// IoU on corner boxes. Written with explicit IEEE intrinsics so pass A and
// pass B produce bit-identical values (required for the is_best equality).
__device__ __forceinline__ float iou_c(float ax0, float ay0, float ax1, float ay1, float aarea,
                                       float gx0, float gy0, float gx1, float gy1, float garea) {
    float ltx = fmaxf(gx0, ax0);
    float lty = fmaxf(gy0, ay0);
    float rbx = fminf(gx1, ax1);
    float rby = fminf(gy1, ay1);
    float wx = fmaxf(__fsub_rn(rbx, ltx), 0.0f);
    float wy = fmaxf(__fsub_rn(rby, lty), 0.0f);
    float inter = __fmul_rn(wx, wy);
    float uni = __fsub_rn(__fadd_rn(garea, aarea), inter);
    return __fdiv_rn(inter, uni);
}

__global__ void init_ws(unsigned* __restrict__ p, int n) {
    int i = blockIdx.x * blockDim.x + threadIdx.x;
    if (i < n) p[i] = 0u;
}

// Pass A: per-gt global max IoU over all anchors (float bits, monotone as uint).
__global__ void __launch_bounds__(256) pass_a(const float* __restrict__ anchors,
                                              const float* __restrict__ gt_boxes,
                                              unsigned* __restrict__ best_bits,
                                              int A, int G) {
    __shared__ float g5[MAXG * 5];
    __shared__ unsigned gmax[MAXG];
    int b = blockIdx.y;
    int tid = threadIdx.x;
    if (tid < G) {
        const float* gp = gt_boxes + ((size_t)b * G + tid) * 4;
        float x0 = gp[0], y0 = gp[1];
        float x1 = __fadd_rn(x0, gp[2]);
        float y1 = __fadd_rn(y0, gp[3]);
        g5[tid * 5 + 0] = x0; g5[tid * 5 + 1] = y0;
        g5[tid * 5 + 2] = x1; g5[tid * 5 + 3] = y1;
        g5[tid * 5 + 4] = __fmul_rn(__fsub_rn(x1, x0), __fsub_rn(y1, y0));
        gmax[tid] = 0u;
    }
    __syncthreads();

    int a = blockIdx.x * 256 + tid;
    int ac = (a < A) ? a : (A - 1);   // clamp: duplicates can't change a max
    const float* ap = anchors + (size_t)ac * 4;
    float ax0 = ap[0], ay0 = ap[1];
    float ax1 = __fadd_rn(ax0, ap[2]);
    float ay1 = __fadd_rn(ay0, ap[3]);
    float aarea = __fmul_rn(__fsub_rn(ax1, ax0), __fsub_rn(ay1, ay0));

    int lane = tid & 31;
    for (int g = 0; g < G; ++g) {
        float iou = iou_c(ax0, ay0, ax1, ay1, aarea,
                          g5[g * 5 + 0], g5[g * 5 + 1], g5[g * 5 + 2], g5[g * 5 + 3], g5[g * 5 + 4]);
        for (int off = 1; off < 32; off <<= 1)
            iou = fmaxf(iou, __shfl_xor(iou, off, 32));
        if (lane == 0) atomicMax(&gmax[g], __float_as_uint(iou));
    }
    __syncthreads();
    if (tid < G) atomicMax(&best_bits[b * G + tid], gmax[tid]);
}

// Pass B: fused match + SmoothL1 + CE; writes neg CE for mining; WMMA block reduce.
__global__ void __launch_bounds__(256) pass_b(const float* __restrict__ preds_loc,
                                              const float* __restrict__ preds_conf,
                                              const float* __restrict__ anchors,
                                              const float* __restrict__ gt_boxes,
                                              const int* __restrict__ gt_labels,
                                              const unsigned* __restrict__ best_bits,
                                              float* __restrict__ accum,
                                              unsigned* __restrict__ num_pos,
                                              float* __restrict__ neg_ce,
                                              int A, int G, int C) {
    __shared__ float g5[MAXG * 5];
    __shared__ unsigned gbest[MAXG];
    __shared__ int glab[MAXG];
    __shared__ float r_sl1[8], r_ce[8];
    __shared__ unsigned r_np[8];

    int b = blockIdx.y;
    int tid = threadIdx.x;
    if (tid < G) {
        const float* gp = gt_boxes + ((size_t)b * G + tid) * 4;
        float x0 = gp[0], y0 = gp[1];
        float x1 = __fadd_rn(x0, gp[2]);
        float y1 = __fadd_rn(y0, gp[3]);
        g5[tid * 5 + 0] = x0; g5[tid * 5 + 1] = y0;
        g5[tid * 5 + 2] = x1; g5[tid * 5 + 3] = y1;
        g5[tid * 5 + 4] = __fmul_rn(__fsub_rn(x1, x0), __fsub_rn(y1, y0));
        gbest[tid] = best_bits[b * G + tid];
        glab[tid] = gt_labels[b * G + tid];
    }
    __syncthreads();

    int a = blockIdx.x * 256 + tid;
    bool valid = (a < A);
    int ac = valid ? a : (A - 1);
    const float* ap = anchors + (size_t)ac * 4;
    float ax0 = ap[0], ay0 = ap[1];
    float ax1 = __fadd_rn(ax0, ap[2]);
    float ay1 = __fadd_rn(ay0, ap[3]);
    float aarea = __fmul_rn(__fsub_rn(ax1, ax0), __fsub_rn(ay1, ay0));

    float mv = -1.0f;   // argmax with first-index tiebreak (strict >)
    int mi = 0;
    bool isbest = false;
    for (int g = 0; g < G; ++g) {
        float iou = iou_c(ax0, ay0, ax1, ay1, aarea,
                          g5[g * 5 + 0], g5[g * 5 + 1], g5[g * 5 + 2], g5[g * 5 + 3], g5[g * 5 + 4]);
        if (iou > mv) { mv = iou; mi = g; }
        isbest |= (__float_as_uint(iou) == gbest[g]);
    }
    bool pos = valid && (isbest || mv >= HIGH_THR_F);
    int label = pos ? glab[mi] : 0;

    // cross entropy for this anchor
    const float* row = preds_conf + ((size_t)b * A + (size_t)ac) * (size_t)C;
    float mx = row[0];
    for (int j = 1; j < C; ++j) mx = fmaxf(mx, row[j]);
    float se = 0.0f;
    for (int j = 0; j < C; ++j) se += expf(row[j] - mx);
    float ce = mx + logf(se) - row[label];

    float ce_pos_p = pos ? ce : 0.0f;
    if (valid) neg_ce[(size_t)b * A + a] = pos ? 0.0f : ce;

    // localization loss for positives
    float sl1p = 0.0f;
    if (pos) {
        const float* lp = preds_loc + ((size_t)b * A + (size_t)a) * 4;
        float mx0 = g5[mi * 5 + 0], my0 = g5[mi * 5 + 1];
        float mx1 = g5[mi * 5 + 2], my1 = g5[mi * 5 + 3];
        float acx = (ax0 + ax1) * 0.5f, acy = (ay0 + ay1) * 0.5f;
        float aw = ax1 - ax0, ah = ay1 - ay0;
        float mcx = (mx0 + mx1) * 0.5f, mcy = (my0 + my1) * 0.5f;
        float mw = mx1 - mx0, mh = my1 - my0;
        float tt0 = (mcx - acx) / (0.1f * aw);
        float tt1 = (mcy - acy) / (0.1f * ah);
        float tt2 = logf(mw / aw) / 0.2f;
        float tt3 = logf(mh / ah) / 0.2f;
        float t[4] = {tt0, tt1, tt2, tt3};
        for (int k = 0; k < 4; ++k) {
            float d = lp[k] - t[k];
            float ad = fabsf(d);
            sl1p += (ad < 1.0f) ? 0.5f * d * d : ad - 0.5f;
        }
    }

    // block reduction: wave sums via WMMA (EXEC all-1s here: no divergence)
    float wsl = wave_sum_wmma(sl1p);
    float wce = wave_sum_wmma(ce_pos_p);
    unsigned long long bal = __ballot(pos);
    int w = tid >> 5;
    int lane = tid & 31;
    if (lane == 0) { r_sl1[w] = wsl; r_ce[w] = wce; r_np[w] = (unsigned)__popcll(bal); }
    __syncthreads();
    if (tid == 0) {
        float s = 0.0f, cc = 0.0f;
        unsigned np = 0;
        int nw = blockDim.x >> 5;
        for (int i = 0; i < nw; ++i) { s += r_sl1[i]; cc += r_ce[i]; np += r_np[i]; }
        atomicAdd(&accum[0], s);
        atomicAdd(&accum[1], cc);
        atomicAdd(&num_pos[b], np);
    }
}

// Pass C: exact top-K sum per image via 8-bit-per-round radix select on float
// bits (all values >= 0 so uint ordering == float ordering). Ties handled
// exactly: leftover count at the threshold value contributes Krem * t.
__global__ void __launch_bounds__(256) select_topk(const float* __restrict__ neg_ce,
                                                   const unsigned* __restrict__ num_pos,
                                                   float* __restrict__ accum, int A) {
    __shared__ unsigned hcnt[256];
    __shared__ float hsum[256];
    __shared__ unsigned s_prefix, s_krem;
    __shared__ float s_acc;
    __shared__ int s_done;

    int b = blockIdx.x;
    int tid = threadIdx.x;
    const float* neg = neg_ce + (size_t)b * A;

    if (tid == 0) {
        unsigned K = (unsigned)NEGPOS * num_pos[b];
        if (K > (unsigned)A) K = (unsigned)A;
        s_krem = K; s_acc = 0.0f; s_prefix = 0u; s_done = (K == 0u) ? 1 : 0;
    }
    __syncthreads();

    for (int r = 0; r < 4; ++r) {
        hcnt[tid] = 0u;
        hsum[tid] = 0.0f;
        __syncthreads();
        if (!s_done) {
            int shift = 24 - 8 * r;
            unsigned pfx = s_prefix;
            for (int i = tid; i < A; i += 256) {
                float v = neg[i];
                unsigned u = __float_as_uint(v);
                if (r == 0 || (u >> (shift + 8)) == pfx) {
                    unsigned bin = (u >> shift) & 255u;
                    atomicAdd(&hcnt[bin], 1u);
                    atomicAdd(&hsum[bin], v);
                }
            }
        }
        __syncthreads();
        if (tid == 0 && !s_done) {
            unsigned krem = s_krem;
            float acc = s_acc;
            unsigned pfx = s_prefix;
            for (int bin = 255; bin >= 0; --bin) {
                unsigned c = hcnt[bin];
                if (c <= krem) {
                    acc += hsum[bin];
                    krem -= c;
                    if (krem == 0u) { s_done = 1; break; }
                } else {
                    pfx = (pfx << 8) | (unsigned)bin;
                    break;
                }
            }
            s_krem = krem; s_acc = acc; s_prefix = pfx;
        }
        __syncthreads();
    }
    if (tid == 0) {
        float acc = s_acc;
        if (!s_done) acc += (float)s_krem * __uint_as_float(s_prefix);
        atomicAdd(&accum[2], acc);
    }
}

__global__ void finalize(const float* __restrict__ accum,
                         const unsigned* __restrict__ num_pos,
                         int B, float* __restrict__ out) {
    if (blockIdx.x == 0 && threadIdx.x == 0) {
        unsigned n = 0;
        for (int b = 0; b < B; ++b) n += num_pos[b];
        float N = (float)n;
        out[0] = accum[0] / (4.0f * N);           // loss_loc
        out[1] = (accum[1] + accum[2]) / N;       // loss_conf
    }
}

extern "C" void kernel_launch(void* const* d_in, const int* in_sizes, int n_in,
                              void* d_out, int out_size, void* d_ws, size_t ws_size,
                              hipStream_t stream) {
    const float* preds_loc  = (const float*)d_in[0];
    const float* preds_conf = (const float*)d_in[1];
    const float* anchors    = (const float*)d_in[2];
    const float* gt_boxes   = (const float*)d_in[3];
    const int*   gt_labels  = (const int*)d_in[4];

    int A = in_sizes[2] / 4;                                    // 131072
    int B = in_sizes[0] / (A * 4);                              // 16
    int C = (int)((long long)in_sizes[1] / ((long long)B * A)); // 21
    int G = in_sizes[4] / B;                                    // 64

    // workspace layout (uint granularity):
    // [0..3] float accum {sl1_sum, ce_pos_sum, hard_neg_sum, pad}
    // [4..4+B) uint num_pos
    // [4+B..4+B+B*G) uint best_bits
    // then: float neg_ce[B*A]  (~8 MB)
    unsigned* ws = (unsigned*)d_ws;
    float*    accum     = (float*)ws;
    unsigned* num_pos   = ws + 4;
    unsigned* best_bits = ws + 4 + B;
    float*    neg       = (float*)(ws + 4 + B + (size_t)B * G);

    int ninit = 4 + B + B * G;
    init_ws<<<(ninit + 255) / 256, 256, 0, stream>>>(ws, ninit);

    dim3 grid((A + 255) / 256, B);
    pass_a<<<grid, 256, 0, stream>>>(anchors, gt_boxes, best_bits, A, G);
    pass_b<<<grid, 256, 0, stream>>>(preds_loc, preds_conf, anchors, gt_boxes, gt_labels,
                                     best_bits, accum, num_pos, neg, A, G, C);
    select_topk<<<B, 256, 0, stream>>>(neg, num_pos, accum, A);
    finalize<<<1, 32, 0, stream>>>(accum, num_pos, B, (float*)d_out);
}